// VectorQuantizer_13417477833224
// MI455X (gfx1250) — compile-verified
//
#include <hip/hip_runtime.h>
#include <hip/hip_bf16.h>

typedef __attribute__((ext_vector_type(2))) float v2f;
typedef __attribute__((ext_vector_type(8))) float v8f;

#define NUM_CODES 512
#define LATENT    256
#define BATCH     64
#define HW        1024
#define NROWS     (BATCH * HW)        // 65536 flattened latent vectors
#define M_TILE    256                 // rows per workgroup (8 waves x 32 rows)
#define N_TILE    64                  // codes per LDS stage
#define ESTR      260                 // padded LDS row stride (even -> 8B-aligned b64)

#define ZQ_ELEMS  (BATCH * LATENT * HW)  // 16777216

// LDS carve-out (floats): eB[64][260] + cnorm[512] + ridx[256]
#define SMEM_FLOATS (N_TILE * ESTR + NUM_CODES + M_TILE)
#define SMEM_BYTES  (SMEM_FLOATS * 4)

// ---------------------------------------------------------------------------
// Kernel 1: cnorm[k] = 0.5 * ||e_k||^2   (one wave per code)
// ---------------------------------------------------------------------------
__global__ void vq_cnorm_kernel(const float* __restrict__ codebook,
                                float* __restrict__ cnorm) {
    const int lane = threadIdx.x & 31;
    const int wave = threadIdx.x >> 5;
    const int code = blockIdx.x * 8 + wave;          // 64 blocks * 8 waves = 512
    const float* row = codebook + code * LATENT;
    float s = 0.0f;
    #pragma unroll
    for (int i = 0; i < LATENT / 32; ++i) {
        float v = row[lane + i * 32];
        s += v * v;
    }
    #pragma unroll
    for (int off = 16; off >= 1; off >>= 1)
        s += __shfl_xor(s, off, 32);
    if (lane == 0) cnorm[code] = 0.5f * s;
}

// ---------------------------------------------------------------------------
// Kernel 2: fused GEMM (f32 WMMA) + argmax + gather/transpose store
//   - A (z rows) fed straight from global: NCHW layout already has hw
//     contiguous, matching the 16x4 A-fragment lane mapping (no transpose).
//   - B (codebook tile) staged in LDS, shared by 8 waves, reused by 2 M-tiles.
// grid = NROWS / M_TILE = 256 blocks, 256 threads (8 waves x 32 rows)
// ---------------------------------------------------------------------------
__global__ void vq_main_kernel(const float* __restrict__ z,
                               const float* __restrict__ codebook,
                               const float* __restrict__ cnorm,
                               float* __restrict__ zq_out,
                               int* __restrict__ idx_out) {
    extern __shared__ float smem[];
    float* eB      = smem;                              // [N_TILE][ESTR]
    float* cnorm_s = eB + N_TILE * ESTR;                // [512]
    int*   ridx    = (int*)(cnorm_s + NUM_CODES);       // [256]

    const int tid  = threadIdx.x;
    const int wave = tid >> 5;
    const int lane = tid & 31;

    const int row0 = blockIdx.x * M_TILE;               // flat row base
    const int b    = row0 >> 10;                        // block never straddles images
    const int hw0  = row0 & (HW - 1);

    // stage 0.5*||e||^2 table
    cnorm_s[tid]       = cnorm[tid];
    cnorm_s[tid + 256] = cnorm[tid + 256];

    // per-lane running best for 2 M-tiles x 8 rows held in the D layout
    float best[2][8];
    int   bidx[2][8];
    #pragma unroll
    for (int mt = 0; mt < 2; ++mt)
        #pragma unroll
        for (int v = 0; v < 8; ++v) { best[mt][v] = -3.4e38f; bidx[mt][v] = 0; }

    const int m    = lane & 15;            // A row / B col / D col within tile
    const int kofs = (lane >> 4) << 1;     // lanes 0-15: K0,K1 ; 16-31: K2,K3

    // A source pointers: z[b][k'][hw] with hw = hw0 + wave*32 + mt*16 + m
    const float* zb  = z + ((size_t)b * LATENT << 10);
    const float* pA0 = zb + hw0 + wave * 32 + m;        // M-tile 0
    const float* pA1 = pA0 + 16;                        // M-tile 1
    const float* brow0 = &eB[m * ESTR + kofs];

    for (int n0 = 0; n0 < NUM_CODES; n0 += N_TILE) {
        __syncthreads();   // previous tile's consumers done before overwrite
        // stage 64 codebook rows (coalesced: one row per iteration)
        #pragma unroll 4
        for (int i = 0; i < N_TILE; ++i)
            eB[i * ESTR + tid] = codebook[(n0 + i) * LATENT + tid];
        __syncthreads();

        v8f acc[2][4];
        #pragma unroll
        for (int mt = 0; mt < 2; ++mt)
            #pragma unroll
            for (int t = 0; t < 4; ++t) acc[mt][t] = (v8f)0.0f;

        // K loop: per step 8x V_WMMA_F32_16X16X4_F32,
        // A: 4 global b32 (L0-cached, coalesced), B: 4 LDS b64 shared by mt
        for (int k = 0; k < LATENT; k += 4) {
            const int ko = (k + kofs) << 10;            // plane offset in floats
            v2f a0, a1;
            a0.x = pA0[ko];  a0.y = pA0[ko + 1024];
            a1.x = pA1[ko];  a1.y = pA1[ko + 1024];
            #pragma unroll
            for (int t = 0; t < 4; ++t) {
                v2f bb = *(const v2f*)(brow0 + t * 16 * ESTR + k);
                acc[0][t] = __builtin_amdgcn_wmma_f32_16x16x4_f32(
                    false, a0, false, bb, (short)0, acc[0][t], false, false);
                acc[1][t] = __builtin_amdgcn_wmma_f32_16x16x4_f32(
                    false, a1, false, bb, (short)0, acc[1][t], false, false);
            }
        }

        // fold into running argmax of (z.e - 0.5||e||^2); strict '>' keeps first min
        #pragma unroll
        for (int t = 0; t < 4; ++t) {
            int   n  = n0 + t * 16 + m;
            float cn = cnorm_s[n];
            #pragma unroll
            for (int mt = 0; mt < 2; ++mt)
                #pragma unroll
                for (int v = 0; v < 8; ++v) {
                    float s = acc[mt][t][v] - cn;
                    if (s > best[mt][v]) { best[mt][v] = s; bidx[mt][v] = n; }
                }
        }
    }

    // reduce across the 16 lanes sharing each M-row set (xor<16 stays in-half)
    #pragma unroll
    for (int off = 8; off >= 1; off >>= 1) {
        #pragma unroll
        for (int mt = 0; mt < 2; ++mt)
            #pragma unroll
            for (int v = 0; v < 8; ++v) {
                float os = __shfl_xor(best[mt][v], off, 32);
                int   oi = __shfl_xor(bidx[mt][v], off, 32);
                if (os > best[mt][v] || (os == best[mt][v] && oi < bidx[mt][v])) {
                    best[mt][v] = os; bidx[mt][v] = oi;
                }
            }
    }
    if (lane == 0 || lane == 16) {
        #pragma unroll
        for (int mt = 0; mt < 2; ++mt) {
            int base = wave * 32 + mt * 16 + ((lane >> 4) << 3);  // rows 0-7 / 8-15
            #pragma unroll
            for (int v = 0; v < 8; ++v) ridx[base + v] = bidx[mt][v];
        }
    }
    __syncthreads();

    // indices out
    idx_out[row0 + tid] = ridx[tid];

    // z_q out: out[b][c][hw0+hwl] = codebook[ridx[hwl]][c]  (coalesced over hwl)
    float* ob = zq_out + ((size_t)b * LATENT << 10);
    #pragma unroll 4
    for (int i = 0; i < M_TILE; ++i) {
        int linear = i * 256 + tid;
        int c   = linear >> 8;        // 0..255
        int hwl = linear & 255;       // 0..255
        ob[(c << 10) + hw0 + hwl] = codebook[ridx[hwl] * LATENT + c];
    }
}

// ---------------------------------------------------------------------------
extern "C" void kernel_launch(void* const* d_in, const int* in_sizes, int n_in,
                              void* d_out, int out_size, void* d_ws, size_t ws_size,
                              hipStream_t stream) {
    const float* z        = (const float*)d_in[0];
    const float* codebook = (const float*)d_in[1];

    float* zq_out  = (float*)d_out;
    int*   idx_out = (int*)(zq_out + ZQ_ELEMS);
    float* cnorm   = (float*)d_ws;                  // 512 floats of scratch

    vq_cnorm_kernel<<<NUM_CODES / 8, 256, 0, stream>>>(codebook, cnorm);
    vq_main_kernel<<<NROWS / M_TILE, 256, SMEM_BYTES, stream>>>(
        z, codebook, cnorm, zq_out, idx_out);
}